// BayesianEncoder_62405874810900
// MI455X (gfx1250) — compile-verified
//
#include <hip/hip_runtime.h>
#include <hip/hip_bf16.h>
#include <math.h>

// ---------------------------------------------------------------------------
// Layout: activations stored transposed as [channel][BPAD] with BPAD = 208
// (batch 200 padded to 208 = 13*16, multiple of 16 for WMMA).
// Padding columns are kept at exactly 0.0f so WMMA batch-sums stay correct.
// ---------------------------------------------------------------------------
#define BPAD 208
#define BREAL 200

typedef __attribute__((ext_vector_type(2))) float v2f;
typedef __attribute__((ext_vector_type(8))) float v8f;

// ---- CDNA5 async global->LDS path (gate on builtin availability) ----------
#if defined(__has_builtin)
#if __has_builtin(__builtin_amdgcn_global_load_async_to_lds_b32)
#define USE_ASYNC_LDS 1
#endif
#if __has_builtin(__builtin_amdgcn_s_wait_asynccnt)
#define HAVE_WAIT_ASYNC_BUILTIN 1
#endif
#endif

#if defined(USE_ASYNC_LDS)
typedef __attribute__((address_space(1))) int gas_int;   // global AS pointer
typedef __attribute__((address_space(3))) int las_int;   // LDS AS pointer

__device__ __forceinline__ void wait_asynccnt0() {
#if defined(HAVE_WAIT_ASYNC_BUILTIN)
    __builtin_amdgcn_s_wait_asynccnt(0);
#else
    asm volatile("s_wait_asynccnt 0x0" ::: "memory");
#endif
}
#endif

// ---------------------------------------------------------------------------
// 1) Transpose x (B, 65536, 3) -> xt [c = node*3+d][BPAD], zero padding cols.
//    On CDNA5 the transposed LDS-tile scatter is done by the async DMA path
//    (global_load_async_to_lds_b32 writes the per-lane LDS address directly).
// ---------------------------------------------------------------------------
__global__ __launch_bounds__(256) void transpose_x_kernel(
    const float* __restrict__ x, float* __restrict__ xt, int C)
{
    __shared__ float tile[32][33];
    const int c0 = blockIdx.x * 32;
    const int tx = threadIdx.x & 31;   // fast index
    const int ty = threadIdx.x >> 5;   // 0..7
    for (int bc = 0; bc < BPAD; bc += 32) {
        for (int r = 0; r < 32; r += 8) {
            int b = bc + r + ty;
            int c = c0 + tx;
#if defined(USE_ASYNC_LDS)
            if (b < BREAL) {
                __builtin_amdgcn_global_load_async_to_lds_b32(
                    (gas_int*)(x + (size_t)b * (size_t)C + c),
                    (las_int*)&tile[tx][r + ty],
                    0, 0);
            } else {
                tile[tx][r + ty] = 0.0f;   // DS store, disjoint addresses
            }
#else
            float v = (b < BREAL) ? x[(size_t)b * (size_t)C + c] : 0.0f;
            tile[tx][r + ty] = v;           // tile[channel_local][b_local]
#endif
        }
#if defined(USE_ASYNC_LDS)
        wait_asynccnt0();                   // this wave's async LDS writes done
#endif
        __syncthreads();
        for (int r = 0; r < 32; r += 8) {
            int c = c0 + r + ty;
            if (c < C) xt[(size_t)c * BPAD + bc + tx] = tile[r + ty][tx];
        }
        __syncthreads();
    }
}

// ---------------------------------------------------------------------------
// 2) Sparse segment pooling.  dst is sorted, so node n's edges form a
//    contiguous segment found by binary search.  One block per output node,
//    lanes cover the (contiguous!) batch axis -> fully coalesced gathers.
// ---------------------------------------------------------------------------
template<int DIN, bool TO_OUT>
__global__ __launch_bounds__(256) void pool_kernel(
    const float* __restrict__ hin,   // [c][BPAD], c = node*DIN + d
    const int*   __restrict__ src,
    const int*   __restrict__ dst,
    const float* __restrict__ w,     // [E][DIN]
    const float* __restrict__ bias,  // [n_out]
    float*       __restrict__ out,   // [n][BPAD]  or  d_out (B,64) if TO_OUT
    int E, int n_out)
{
    const int n = blockIdx.x;
    __shared__ int s_lo, s_hi;
    if (threadIdx.x == 0) {
        int lo = 0, hi = E;
        while (lo < hi) { int mid = (lo + hi) >> 1; if (dst[mid] < n) lo = mid + 1; else hi = mid; }
        s_lo = lo;
        int lo2 = lo, hi2 = E;
        while (lo2 < hi2) { int mid = (lo2 + hi2) >> 1; if (dst[mid] < n + 1) lo2 = mid + 1; else hi2 = mid; }
        s_hi = lo2;
    }
    __syncthreads();
    const int b = threadIdx.x;
    if (b >= BPAD) return;

    const int lo = s_lo, hi = s_hi;
    float acc = 0.0f;
    for (int e = lo; e < hi; ++e) {
        int s = src[e];
        if (e + 1 < hi) {
            // CDNA5 global_prefetch_b8 for the next gathered row
            __builtin_prefetch(&hin[(size_t)src[e + 1] * DIN * BPAD + b], 0, 0);
        }
        if (DIN == 3) {
            const float* hp = hin + (size_t)s * 3 * BPAD + b;
            acc = fmaf(hp[0 * BPAD], w[e * 3 + 0], acc);
            acc = fmaf(hp[1 * BPAD], w[e * 3 + 1], acc);
            acc = fmaf(hp[2 * BPAD], w[e * 3 + 2], acc);
        } else {
            acc = fmaf(hin[(size_t)s * BPAD + b], w[e], acc);
        }
    }
    if (TO_OUT) {
        if (b < BREAL) out[(size_t)b * n_out + n] = acc + bias[n];
    } else {
        if (b < BREAL) out[(size_t)n * BPAD + b] = acc + bias[n];
        else           out[(size_t)n * BPAD + b] = 0.0f;   // keep padding zero
    }
}

// ---------------------------------------------------------------------------
// 3) Fused BatchNorm(batch stats) + ReLU + dropout.  One wave per 16-node
//    tile; batch sums & sums-of-squares via V_WMMA_F32_16X16X4_F32 with an
//    all-ones A matrix (D[0][j] = column sums -> exact f32 batch reduction).
// ---------------------------------------------------------------------------
__global__ __launch_bounds__(32) void bn_relu_drop_kernel(
    float* __restrict__ h,                 // [n][BPAD], padding cols == 0
    const float* __restrict__ gamma,
    const float* __restrict__ beta,
    unsigned seed)
{
    const int n0   = blockIdx.x * 16;
    const int lane = threadIdx.x;
    const int j    = lane & 15;
    const int kh   = lane >> 4;
    const float* base = h + (size_t)(n0 + j) * BPAD + kh;

    v8f sum = {0.f,0.f,0.f,0.f,0.f,0.f,0.f,0.f};
    v8f ssq = {0.f,0.f,0.f,0.f,0.f,0.f,0.f,0.f};
    const v2f A = {1.0f, 1.0f};

    for (int bc = 0; bc < BPAD; bc += 4) {     // 52 chunks of 4 batch rows
        float b0 = base[bc];
        float b1 = base[bc + 2];
        v2f Bv = {b0, b1};
        v2f B2 = {b0 * b0, b1 * b1};
        sum = __builtin_amdgcn_wmma_f32_16x16x4_f32(false, A, false, Bv,
                                                    (short)0, sum, false, false);
        ssq = __builtin_amdgcn_wmma_f32_16x16x4_f32(false, A, false, B2,
                                                    (short)0, ssq, false, false);
    }

    __shared__ float s_scale[16], s_shift[16];
    if (lane < 16) {
        float m  = sum[0] * (1.0f / (float)BREAL);
        float vr = ssq[0] * (1.0f / (float)BREAL) - m * m;   // biased var
        float rs = rsqrtf(vr + 1e-5f);
        float g  = gamma[n0 + lane];
        float bt = beta[n0 + lane];
        s_scale[lane] = rs * g;
        s_shift[lane] = bt - m * rs * g;
    }
    __syncthreads();

    for (int jj = 0; jj < 16; ++jj) {
        float sc = s_scale[jj], sh = s_shift[jj];
        float* row = h + (size_t)(n0 + jj) * BPAD;
        for (int b = lane; b < BREAL; b += 32) {
            float v = fmaxf(row[b] * sc + sh, 0.0f);
            // deterministic counter-based dropout mask (p_drop = 0.1)
            unsigned id = ((unsigned)(n0 + jj) * 20011u + (unsigned)b) ^ seed;
            unsigned s  = id * 747796405u + 2891336453u;
            unsigned wd = ((s >> ((s >> 28) + 4u)) ^ s) * 277803737u;
            unsigned r  = (wd >> 22) ^ wd;
            v = (r < 3865470566u) ? v * (1.0f / 0.9f) : 0.0f;  // 0.9 * 2^32
            row[b] = v;
        }
        // padding columns stay exactly 0 for the next layer's WMMA sums
    }
}

// ---------------------------------------------------------------------------
// 4) KL = 0.5 * sum(w^2) + 999.5 * count  (deterministic 2-stage reduction)
// ---------------------------------------------------------------------------
__global__ __launch_bounds__(256) void sq_partial_kernel(
    const float* __restrict__ w, int n, float* __restrict__ partial)
{
    __shared__ float red[256];
    float acc = 0.0f;
    for (int i = blockIdx.x * 256 + threadIdx.x; i < n; i += gridDim.x * 256) {
        float v = w[i];
        acc = fmaf(v, v, acc);
    }
    red[threadIdx.x] = acc;
    __syncthreads();
    for (int s = 128; s > 0; s >>= 1) {
        if (threadIdx.x < s) red[threadIdx.x] += red[threadIdx.x + s];
        __syncthreads();
    }
    if (threadIdx.x == 0) partial[blockIdx.x] = red[0];
}

__global__ void kl_finalize_kernel(const float* __restrict__ partials, int n,
                                   float constTerm, float* __restrict__ out)
{
    if (threadIdx.x == 0 && blockIdx.x == 0) {
        float s = 0.0f;
        for (int i = 0; i < n; ++i) s += partials[i];
        out[0] = 0.5f * s + constTerm;
    }
}

// ---------------------------------------------------------------------------
// Host launcher
// ---------------------------------------------------------------------------
extern "C" void kernel_launch(void* const* d_in, const int* in_sizes, int n_in,
                              void* d_out, int out_size, void* d_ws, size_t ws_size,
                              hipStream_t stream)
{
    (void)n_in; (void)ws_size;
    const float* x = (const float*)d_in[0];

    const int*   src[5]; const int* dst[5];
    const float* w[5];   const float* bias[5];
    const float* gam[4]; const float* bet[4];
    int E[5], WN[5], NOUT[5];
    int idx = 1;
    for (int l = 0; l < 5; ++l) {
        src[l]  = (const int*)d_in[idx];   E[l]  = in_sizes[idx]; idx++;
        dst[l]  = (const int*)d_in[idx];   idx++;
        w[l]    = (const float*)d_in[idx]; WN[l] = in_sizes[idx]; idx++;
        bias[l] = (const float*)d_in[idx]; NOUT[l] = in_sizes[idx]; idx++;
        if (l < 4) {
            gam[l] = (const float*)d_in[idx]; idx++;
            bet[l] = (const float*)d_in[idx]; idx++;
        }
    }

    // workspace layout (floats)
    float* ws = (float*)d_ws;
    const size_t CH = 65536UL * 3UL;                  // 196608 input channels
    float* xt = ws;                size_t off = CH * BPAD;
    float* hbuf[4];
    for (int l = 0; l < 4; ++l) { hbuf[l] = ws + off; off += (size_t)NOUT[l] * BPAD; }
    float* klp = ws + off;                            // 5 * 64 partials

    // 1) transpose x into padded channel-major layout
    transpose_x_kernel<<<(int)(CH / 32), 256, 0, stream>>>(x, xt, (int)CH);

    // 2) layer 1 (din = 3)
    pool_kernel<3, false><<<NOUT[0], 256, 0, stream>>>(
        xt, src[0], dst[0], w[0], bias[0], hbuf[0], E[0], NOUT[0]);
    bn_relu_drop_kernel<<<NOUT[0] / 16, 32, 0, stream>>>(hbuf[0], gam[0], bet[0],
                                                         0x9E3779B9u * 1u);
    // 3) layers 2..4 (din = 1)
    for (int l = 1; l < 4; ++l) {
        pool_kernel<1, false><<<NOUT[l], 256, 0, stream>>>(
            hbuf[l - 1], src[l], dst[l], w[l], bias[l], hbuf[l], E[l], NOUT[l]);
        bn_relu_drop_kernel<<<NOUT[l] / 16, 32, 0, stream>>>(hbuf[l], gam[l], bet[l],
                                                             0x9E3779B9u * (unsigned)(l + 1));
    }
    // 4) layer 5 -> d_out (B, 64), row-major
    pool_kernel<1, true><<<NOUT[4], 256, 0, stream>>>(
        hbuf[3], src[4], dst[4], w[4], bias[4], (float*)d_out, E[4], NOUT[4]);

    // 5) KL scalar -> d_out[out_size - 1]
    long long total = 0;
    for (int l = 0; l < 5; ++l) total += WN[l];
    const float constTerm = 999.5f * (float)total;    // 0.5*(w^2-1) + 1000 per weight
    for (int l = 0; l < 5; ++l)
        sq_partial_kernel<<<64, 256, 0, stream>>>(w[l], WN[l], klp + 64 * l);
    kl_finalize_kernel<<<1, 32, 0, stream>>>(klp, 320, constTerm,
                                             (float*)d_out + (out_size - 1));
}